// FedTGPClientLoss_82892868813506
// MI455X (gfx1250) — compile-verified
//
#include <hip/hip_runtime.h>
#include <hip/hip_bf16.h>
#include <math.h>

// Problem sizes from setup_inputs()
#define BB 16384   // batch
#define CC 1000    // classes
#define DD 512     // feature dim

typedef __attribute__((ext_vector_type(2))) float v2f;
typedef __attribute__((ext_vector_type(8))) float v8f;

// ---------------------------------------------------------------------------
// Kernel 1: cross-entropy partials. One block (256 thr = 8 wave32) per row.
// Threads 0..249 each load one float4 (b128) -> whole 1000-elem row in regs,
// single pass over HBM. max -> sum(exp) -> gather at label.
// ---------------------------------------------------------------------------
__global__ void __launch_bounds__(256) ce_kernel(const float* __restrict__ logits,
                                                 const int* __restrict__ labels,
                                                 float* __restrict__ ce_part) {
    const int row = blockIdx.x;
    const int t   = threadIdx.x;
    const float* lr = logits + (size_t)row * CC;

    const bool vld = (t < CC / 4);          // 250 threads carry data
    float4 xv = make_float4(-INFINITY, -INFINITY, -INFINITY, -INFINITY);
    if (vld) xv = ((const float4*)lr)[t];   // global_load_b128, coalesced

    __shared__ float red[8];
    __shared__ float bcast;
    const int lane = t & 31;
    const int wid  = t >> 5;

    // ---- block max ----
    float m = fmaxf(fmaxf(xv.x, xv.y), fmaxf(xv.z, xv.w));
#pragma unroll
    for (int off = 16; off >= 1; off >>= 1)
        m = fmaxf(m, __shfl_xor(m, off, 32));
    if (lane == 0) red[wid] = m;
    __syncthreads();
    if (t == 0) {
        float mm = red[0];
#pragma unroll
        for (int i = 1; i < 8; ++i) mm = fmaxf(mm, red[i]);
        bcast = mm;
    }
    __syncthreads();
    m = bcast;

    // ---- block sum of exp(x - m) ----
    float s = 0.0f;
    if (vld) {
        s = __expf(xv.x - m) + __expf(xv.y - m) +
            __expf(xv.z - m) + __expf(xv.w - m);
    }
#pragma unroll
    for (int off = 16; off >= 1; off >>= 1)
        s += __shfl_xor(s, off, 32);
    if (lane == 0) red[wid] = s;
    __syncthreads();
    if (t == 0) {
        float tot = 0.0f;
#pragma unroll
        for (int i = 0; i < 8; ++i) tot += red[i];
        const int   lbl = labels[row];
        const float ll  = lr[lbl];                 // row just streamed -> cache hit
        ce_part[row] = -(ll - m - __logf(tot));    // -log_softmax at label
    }
}

// ---------------------------------------------------------------------------
// Kernel 2: prototype-MSE partials. One block (256 thr = 8 waves) handles TWO
// rows: waves 0-3 -> row 2*blk, waves 4-7 -> row 2*blk+1. Each thread loads a
// float4 (b128) of features and of the gathered prototype (protos stay L2-
// resident: 2 MB << 192 MB), pre-sums its 4 squared diffs pairwise into the
// 2-element f32 A fragment, and one V_WMMA_F32_16X16X4_F32 per wave reduces
// the wave's 128 elements:
//   A (16x4 f32) = 32 lanes x 2 VGPRs (all 64 slots), B = all-ones (layout
//   irrelevant: constant), D[m][n] = rowsum_m(A). Documented D layout: lanes
//   0-15 hold rows 0-7 in VGPRs 0-7, lanes 16-31 hold rows 8-15 -> add the 8
//   D VGPRs + shfl_xor(16) = full wave sum. No divergence before the WMMA
//   (row/label selection is arithmetic), so EXEC is all-ones as required.
// ---------------------------------------------------------------------------
__global__ void __launch_bounds__(256) proto_kernel(const float* __restrict__ feats,
                                                    const int* __restrict__ labels,
                                                    const float* __restrict__ protos,
                                                    float* __restrict__ proto_part) {
    const int t    = threadIdx.x;
    const int half = t >> 7;                    // 0: waves 0-3, 1: waves 4-7
    const int lt   = t & 127;                   // thread-in-row
    const int row  = blockIdx.x * 2 + half;
    const int lbl  = labels[row];

    const float4 fv = ((const float4*)(feats  + (size_t)row * DD))[lt];
    const float4 pv = ((const float4*)(protos + (size_t)lbl * DD))[lt];
    const float e0 = fv.x - pv.x;
    const float e1 = fv.y - pv.y;
    const float e2 = fv.z - pv.z;
    const float e3 = fv.w - pv.w;

    v2f a;
    a[0] = e0 * e0 + e1 * e1;
    a[1] = e2 * e2 + e3 * e3;
    v2f ones;
    ones[0] = 1.0f;
    ones[1] = 1.0f;
    v8f c = {};
    // D = A x ones + 0 -> per-row sums of the squared diffs
    v8f d = __builtin_amdgcn_wmma_f32_16x16x4_f32(
        /*neg_a=*/false, a, /*neg_b=*/false, ones,
        /*c_mod=*/(short)0, c, /*reuse_a=*/false, /*reuse_b=*/false);

    float s = d[0] + d[1] + d[2] + d[3] + d[4] + d[5] + d[6] + d[7];
    s += __shfl_xor(s, 16, 32);   // rows 0-7 half + rows 8-15 half

    __shared__ float red[8];
    if ((t & 31) == 0) red[t >> 5] = s;
    __syncthreads();
    if (lt == 0) {                // thread 0 -> row A, thread 128 -> row B
        const int b = half * 4;
        proto_part[row] = red[b] + red[b + 1] + red[b + 2] + red[b + 3];
    }
}

// ---------------------------------------------------------------------------
// Kernel 3: deterministic final reduction of both partial arrays + epilogue.
// Single block, fixed-order strided accumulation + shared-memory tree.
// ---------------------------------------------------------------------------
__global__ void __launch_bounds__(256) finalize_kernel(const float* __restrict__ ce_part,
                                                       const float* __restrict__ proto_part,
                                                       float* __restrict__ out) {
    __shared__ float s1[256];
    __shared__ float s2[256];
    const int t = threadIdx.x;
    float a = 0.0f, b = 0.0f;
#pragma unroll 4
    for (int k = 0; k < BB / 256; ++k) {
        a += ce_part[t + k * 256];
        b += proto_part[t + k * 256];
    }
    s1[t] = a;
    s2[t] = b;
    __syncthreads();
    for (int stride = 128; stride > 0; stride >>= 1) {
        if (t < stride) {
            s1[t] += s1[t + stride];
            s2[t] += s2[t + stride];
        }
        __syncthreads();
    }
    if (t == 0) {
        float ce = s1[0] / (float)BB;
        if (!isfinite(ce)) ce = 0.0f;
        float proto = s2[0] / ((float)BB * (float)DD);
        float total = ce + proto;               // LAMDA = 1.0
        if (!isfinite(total)) total = ce;
        out[0] = total;
        out[1] = ce;
        out[2] = proto;
    }
}

extern "C" void kernel_launch(void* const* d_in, const int* in_sizes, int n_in,
                              void* d_out, int out_size, void* d_ws, size_t ws_size,
                              hipStream_t stream) {
    const float* logits   = (const float*)d_in[0];   // [B, C]
    const int*   labels   = (const int*)d_in[1];     // [B]
    const float* features = (const float*)d_in[2];   // [B, D]
    const float* protos   = (const float*)d_in[3];   // [C, D]
    float*       out      = (float*)d_out;           // [3]: total, ce, proto

    float* ce_part    = (float*)d_ws;                // [B]
    float* proto_part = ce_part + BB;                // [B]

    ce_kernel<<<dim3(BB), dim3(256), 0, stream>>>(logits, labels, ce_part);
    proto_kernel<<<dim3(BB / 2), dim3(256), 0, stream>>>(features, labels, protos, proto_part);
    finalize_kernel<<<dim3(1), dim3(256), 0, stream>>>(ce_part, proto_part, out);
}